// BackwardTransformLayer_59983513256496
// MI455X (gfx1250) — compile-verified
//
#include <hip/hip_runtime.h>

// Inverse DWT (DB4 synthesis, circular) for 4096x4096 f32 inputs -> 4096x8192 f32.
// Memory-bound (AI~2 flop/byte): optimize the data path for MI455X (23.3 TB/s).
// Uses gfx1250 async global->LDS (GLOBAL_LOAD_ASYNC_TO_LDS_B128 / ASYNCcnt) when
// the toolchain exposes the builtins, with a synchronous fallback.

#define N_ROWS  4096
#define M_COLS  4096
#define TILE    1024
#define HALO    4        // need 3, pad to 4 floats (16B) to keep b128 LDS dst aligned
#define THREADS 256

typedef float __attribute__((ext_vector_type(2))) f32x2;
typedef float __attribute__((ext_vector_type(4))) f32x4;
typedef int   __attribute__((ext_vector_type(4))) i32x4;

// Exact parameter types of the async-LDS builtins on this toolchain:
// (int4 addrspace(1)* gsrc, int4 addrspace(3)* ldst, imm offset, imm cpol)
typedef __attribute__((address_space(1))) i32x4* g_i32x4_p;
typedef __attribute__((address_space(3))) i32x4* l_i32x4_p;

#if defined(__has_builtin)
#if __has_builtin(__builtin_amdgcn_global_load_async_to_lds_b128)
#define USE_ASYNC_LDS 1
#endif
#endif

__global__ __launch_bounds__(THREADS) void idwt_db4_rows_kernel(
    const float* __restrict__ D,   // details       [N_ROWS][M_COLS]
    const float* __restrict__ A,   // approximation [N_ROWS][M_COLS]
    const float* __restrict__ wav, // DB4, 8 taps
    float* __restrict__ out)       // [N_ROWS][2*M_COLS]
{
    __shared__ __attribute__((aligned(16))) float sD[HALO + TILE];
    __shared__ __attribute__((aligned(16))) float sA[HALO + TILE];

    const int tid = threadIdx.x;
    const int row = blockIdx.y;
    const int u0  = blockIdx.x * TILE;

    const float* drow = D + (size_t)row * M_COLS;
    const float* arow = A + (size_t)row * M_COLS;

    // ---- main tile: 256 lanes x 16B, contiguous + 16B aligned both sides ----
#ifdef USE_ASYNC_LDS
    __builtin_amdgcn_global_load_async_to_lds_b128(
        (g_i32x4_p)(drow + u0 + tid * 4),
        (l_i32x4_p)&sD[HALO + tid * 4],
        /*offset=*/0, /*cpol=*/0);
    __builtin_amdgcn_global_load_async_to_lds_b128(
        (g_i32x4_p)(arow + u0 + tid * 4),
        (l_i32x4_p)&sA[HALO + tid * 4],
        /*offset=*/0, /*cpol=*/0);
#else
    {
        const f32x4 vd = *(const f32x4*)(drow + u0 + tid * 4);
        const f32x4 va = *(const f32x4*)(arow + u0 + tid * 4);
        *(f32x4*)&sD[HALO + tid * 4] = vd;
        *(f32x4*)&sA[HALO + tid * 4] = va;
    }
#endif

    // ---- circular halo: columns (u0-4 .. u0-1) mod M ----
    if (tid < HALO) {
        const int idx = (u0 - HALO + tid) & (M_COLS - 1);
        sD[tid] = drow[idx];
    } else if (tid < 2 * HALO) {
        const int t   = tid - HALO;
        const int idx = (u0 - HALO + t) & (M_COLS - 1);
        sA[t] = arow[idx];
    }

#ifdef USE_ASYNC_LDS
#if __has_builtin(__builtin_amdgcn_s_wait_asynccnt)
    __builtin_amdgcn_s_wait_asynccnt(0);
#else
    asm volatile("s_wait_asynccnt 0" ::: "memory");
#endif
#endif
    __syncthreads();

    // Filter taps (uniform address -> scalar loads; 8 floats, hits constant path)
    const float w0 = wav[0], w1 = wav[1], w2 = wav[2], w3 = wav[3];
    const float w4 = wav[4], w5 = wav[5], w6 = wav[6], w7 = wav[7];

    float* orow = out + (size_t)row * (2 * M_COLS);

#pragma unroll
    for (int k = 0; k < TILE / THREADS; ++k) {
        const int ul = tid + k * THREADS;    // local column u - u0
        const int b  = HALO + ul;
        const float d0 = sD[b - 3], d1 = sD[b - 2], d2 = sD[b - 1], d3 = sD[b];
        const float a0 = sA[b - 3], a1 = sA[b - 2], a2 = sA[b - 1], a3 = sA[b];

        // out[2u]   = sum_j D[u-j]*flip(w)[2j]   + A[u-j]*(w*sign)[2j]
        // out[2u+1] = sum_j D[u-j]*flip(w)[2j+1] + A[u-j]*(w*sign)[2j+1]
        const float even = d3 * w7 + d2 * w5 + d1 * w3 + d0 * w1
                         + a3 * w0 + a2 * w2 + a1 * w4 + a0 * w6;
        const float odd  = d3 * w6 + d2 * w4 + d1 * w2 + d0 * w0
                         - (a3 * w1 + a2 * w3 + a1 * w5 + a0 * w7);

        f32x2 v; v.x = even; v.y = odd;
        // streamed once, >L2 working set -> non-temporal store
        __builtin_nontemporal_store(v, (f32x2*)(orow + 2 * (u0 + ul)));
    }
}

extern "C" void kernel_launch(void* const* d_in, const int* in_sizes, int n_in,
                              void* d_out, int out_size, void* d_ws, size_t ws_size,
                              hipStream_t stream) {
    (void)in_sizes; (void)n_in; (void)out_size; (void)d_ws; (void)ws_size;
    const float* D   = (const float*)d_in[0];
    const float* A   = (const float*)d_in[1];
    const float* wav = (const float*)d_in[2];
    float* out = (float*)d_out;

    dim3 grid(M_COLS / TILE, N_ROWS, 1);
    idwt_db4_rows_kernel<<<grid, dim3(THREADS, 1, 1), 0, stream>>>(D, A, wav, out);
}